// ST_71725953843680
// MI455X (gfx1250) — compile-verified
//
#include <hip/hip_runtime.h>
#include <math.h>

typedef __attribute__((ext_vector_type(2))) float v2f;
typedef __attribute__((ext_vector_type(8))) float v8f;

// D = A(16x4) * B(4x16) + C, fp32 WMMA (CDNA5)
#define WMMA_F32(a, b, c) \
  __builtin_amdgcn_wmma_f32_16x16x4_f32(false, (a), false, (b), (short)0, (c), false, false)

// ws layout (floats):
//   [0..1023]        lap (32x32)
//   [1024..1151]     csum1: sum_c w1[c*5+0][o]   (k=0 ones-term, layer1)
//   [1152..1279]     csum2: sum_c w2[c*5+0][o]   (k=0 ones-term, layer2)
//   [1280..32000)    wp1: packed w1, [ki][cp][h][2], ki=k-1 (4), cp (30), h (128)
//   [32000..97536)   wp2: packed w2, [ki][cp][h][2], ki (4), cp (64), h (128)
#define WP1_OFF 1280
#define WP2_OFF 32000
#define WS_NEED_FLOATS 97536

// ---------------------------------------------------------------------------
// Stage 0: Laplacian + k=0 column sums (+ optional weight repack).
// ---------------------------------------------------------------------------
template <bool PACK>
__global__ __launch_bounds__(256) void dgcnn_stage0(
    const float* __restrict__ adj, const float* __restrict__ adj_bias,
    const float* __restrict__ w1, const float* __restrict__ w2,
    float* __restrict__ ws) {
  __shared__ float dre[32];
  const int tid = threadIdx.x;
  const float bias = adj_bias[0];

  if (tid < 32) {
    float s = 0.f;
    for (int j = 0; j < 32; ++j) {
      float v = adj[tid * 32 + j] + bias;
      s += (v > 0.f) ? v : 0.f;
    }
    dre[tid] = 1.0f / sqrtf(s + 1e-5f);
  }
  __syncthreads();

  for (int idx = tid; idx < 1024; idx += 256) {
    int i = idx >> 5, j = idx & 31;
    float v = adj[idx] + bias;
    v = (v > 0.f) ? v : 0.f;
    ws[idx] = ((i == j) ? 1.0f : 0.0f) - dre[i] * v * dre[j];
  }

  if (tid < 128) {
    float s = 0.f;
    for (int c = 0; c < 60; ++c) s += w1[(c * 5) * 128 + tid];
    ws[1024 + tid] = s;
  } else {
    int o = tid - 128;
    float s = 0.f;
    for (int c = 0; c < 128; ++c) s += w2[(c * 5) * 128 + o];
    ws[1152 + o] = s;
  }

  if (PACK) {
    // wp1: ki in 0..3 (k=ki+1), cp in 0..29 (c pair 2cp,2cp+1), h in 0..127
    for (int idx = tid; idx < 4 * 30 * 128; idx += 256) {
      int h = idx & 127;
      int rest = idx >> 7;
      int cp = rest % 30;
      int k = rest / 30 + 1;
      int c0 = 2 * cp;
      ws[WP1_OFF + 2 * idx]     = w1[(c0 * 5 + k) * 128 + h];
      ws[WP1_OFF + 2 * idx + 1] = w1[((c0 + 1) * 5 + k) * 128 + h];
    }
    // wp2: ki in 0..3, cp in 0..63, h in 0..127
    for (int idx = tid; idx < 4 * 64 * 128; idx += 256) {
      int h = idx & 127;
      int rest = idx >> 7;
      int cp = rest & 63;
      int k = (rest >> 6) + 1;
      int c0 = 2 * cp;
      ws[WP2_OFF + 2 * idx]     = w2[(c0 * 5 + k) * 128 + h];
      ws[WP2_OFF + 2 * idx + 1] = w2[((c0 + 1) * 5 + k) * 128 + h];
    }
  }
}

// ---------------------------------------------------------------------------
// Chebyshev step (2 stacked batches, 64 rows): dst = 2*(L @ src) - prev,
// or L @ src when !HAS_PREV.  L is block-diagonal per batch.
// CTL = log2(ctiles); ST = 16<<CTL; uniform counted tile loop (no exec
// masking around the WMMAs, full unroll, constant ds offsets).
// ---------------------------------------------------------------------------
template <int CTL, bool HAS_PREV>
__device__ __forceinline__ void cheb_step(const float* __restrict__ lap_s,
                                          const float* __restrict__ src,
                                          const float* __restrict__ prev,
                                          float* __restrict__ dst, int wave,
                                          int lane) {
  constexpr int CTILES = 1 << CTL;
  constexpr int ST = 16 * CTILES;
  constexpr int NT = (4 * CTILES) / 8;  // tiles per wave (uniform)
  const int lrow = lane & 15;
  const int half = lane >> 4;
  const int khalf = half << 1;
#pragma unroll
  for (int i = 0; i < NT; ++i) {
    int t = wave + i * 8;
    int rt = t >> CTL;             // row tile 0..3 (rt>>1 = batch)
    int ct = t & (CTILES - 1);
    int bat = rt >> 1;
    int m0 = (rt & 1) * 16;        // row within the 32x32 Laplacian
    int n0 = ct * 16;
    v8f acc = {};
#pragma unroll
    for (int kk = 0; kk < 32; kk += 4) {
      int kb = kk + khalf;
      v2f a = *(const v2f*)&lap_s[(m0 + lrow) * 32 + kb];  // ds_load_b64
      v2f bf;
      bf.x = src[(bat * 32 + kb) * ST + n0 + lrow];
      bf.y = src[(bat * 32 + kb + 1) * ST + n0 + lrow];
      acc = WMMA_F32(a, bf, acc);
    }
#pragma unroll
    for (int r = 0; r < 8; ++r) {
      int row = rt * 16 + r + half * 8;
      int col = n0 + lrow;
      float v = acc[r];
      dst[row * ST + col] = HAS_PREV ? (2.0f * v - prev[row * ST + col]) : v;
    }
  }
}

// ---------------------------------------------------------------------------
// One Chebyshev-order GEMM pass over 64 rows (4 row-tiles per wave, col-tile
// = wave).  PACKED: w points at wp[k] base ([cp][h][2], single b64/lane).
// Unpacked: w is the raw weight matrix, k its Chebyshev order.
// ---------------------------------------------------------------------------
template <bool PACKED, int ST, int CEFF>
__device__ __forceinline__ void gemm_pass(const float* __restrict__ t,
                                          const float* __restrict__ w, int k,
                                          int wave, int lane,
                                          v8f* __restrict__ acc) {
  const int lrow = lane & 15;
  const int khalf = (lane >> 4) << 1;
  const int h0 = wave * 16 + lrow;
  for (int c = 0; c < CEFF; c += 4) {
    int cb = c + khalf;
    v2f bf;
    if (PACKED) {
      bf = *(const v2f*)&w[((cb >> 1) * 128 + h0) * 2];   // global_load_b64
    } else {
      bf.x = w[(cb * 5 + k) * 128 + h0];
      bf.y = w[((cb + 1) * 5 + k) * 128 + h0];
    }
    v2f a0 = *(const v2f*)&t[lrow * ST + cb];
    v2f a1 = *(const v2f*)&t[(lrow + 16) * ST + cb];
    v2f a2 = *(const v2f*)&t[(lrow + 32) * ST + cb];
    v2f a3 = *(const v2f*)&t[(lrow + 48) * ST + cb];
    acc[0] = WMMA_F32(a0, bf, acc[0]);
    acc[1] = WMMA_F32(a1, bf, acc[1]);
    acc[2] = WMMA_F32(a2, bf, acc[2]);
    acc[3] = WMMA_F32(a3, bf, acc[3]);
  }
}

// ---------------------------------------------------------------------------
// Main fused kernel: TWO batches per 256-thread block (8 wave32, 64 M-rows).
// ---------------------------------------------------------------------------
template <bool PACKED>
__global__ __launch_bounds__(256) void dgcnn_main(
    const float* __restrict__ x, const float* __restrict__ w1,
    const float* __restrict__ b1, const float* __restrict__ w2,
    const float* __restrict__ b2, const float* __restrict__ fc_w,
    const float* __restrict__ fc_b, const float* __restrict__ ws,
    float* __restrict__ out) {
  __shared__ float lap_s[32 * 32];
  __shared__ float tA[64 * 128];
  __shared__ float tB[64 * 128];
  __shared__ float red[1024];

  const int tid = threadIdx.x;
  const int lane = tid & 31;
  const int wave = tid >> 5;
  const int b0 = blockIdx.x * 2;

  for (int i = tid; i < 1024; i += 256) lap_s[i] = ws[i];
  // Stage x[b0], x[b0+1] into tA as T1: 64 rows, stride 64, cols 60..63 = 0.
  for (int i = tid; i < 64 * 64; i += 256) {
    int r = i >> 6, c = i & 63;
    int bat = r >> 5, rb = r & 31;
    tA[r * 64 + c] =
        (c < 60) ? x[(size_t)(b0 + bat) * 1920 + rb * 60 + c] : 0.0f;
  }
  __syncthreads();

  const float* wk1 = PACKED ? (ws + WP1_OFF) : w1;  // per-k stride 7680 packed
  const float* wk2 = PACKED ? (ws + WP2_OFF) : w2;  // per-k stride 16384 packed
  const int s1 = PACKED ? 7680 : 0;
  const int s2 = PACKED ? 16384 : 0;

  // ---------------- Layer 1 (C=60 padded to 64, H=128) ----------------
  v8f acc[4] = {};
  cheb_step<2, false>(lap_s, tA, nullptr, tB, wave, lane);          // T2=L@x
  __syncthreads();
  gemm_pass<PACKED, 64, 60>(tA, wk1 + 0 * s1, 1, wave, lane, acc);  // k=1
  gemm_pass<PACKED, 64, 60>(tB, wk1 + 1 * s1, 2, wave, lane, acc);  // k=2
  __syncthreads();
  cheb_step<2, true>(lap_s, tB, tA, tA, wave, lane);                // T3->tA
  __syncthreads();
  gemm_pass<PACKED, 64, 60>(tA, wk1 + 2 * s1, 3, wave, lane, acc);  // k=3
  __syncthreads();
  cheb_step<2, true>(lap_s, tA, tB, tB, wave, lane);                // T4->tB
  __syncthreads();
  gemm_pass<PACKED, 64, 60>(tB, wk1 + 3 * s1, 4, wave, lane, acc);  // k=4
  __syncthreads();  // all reads of tA done; safe to overwrite with h1

  {  // epilogue: h1 = relu(acc + csum1 + b1) -> tA (stride 128)
    int col = wave * 16 + (lane & 15);
    float add = ws[1024 + col] + b1[col];
#pragma unroll
    for (int rt = 0; rt < 4; ++rt) {
#pragma unroll
      for (int r = 0; r < 8; ++r) {
        int row = rt * 16 + r + ((lane >> 4) << 3);
        float v = acc[rt][r] + add;
        tA[row * 128 + col] = (v > 0.f) ? v : 0.f;
      }
    }
  }
  __syncthreads();

  // ---------------- Layer 2 (C=128, H=128) ----------------
  v8f d[4] = {};
  cheb_step<3, false>(lap_s, tA, nullptr, tB, wave, lane);          // T2=L@h1
  __syncthreads();
  gemm_pass<PACKED, 128, 128>(tA, wk2 + 0 * s2, 1, wave, lane, d);
  gemm_pass<PACKED, 128, 128>(tB, wk2 + 1 * s2, 2, wave, lane, d);
  __syncthreads();
  cheb_step<3, true>(lap_s, tB, tA, tA, wave, lane);                // T3->tA
  __syncthreads();
  gemm_pass<PACKED, 128, 128>(tA, wk2 + 2 * s2, 3, wave, lane, d);
  __syncthreads();
  cheb_step<3, true>(lap_s, tA, tB, tB, wave, lane);                // T4->tB
  __syncthreads();
  gemm_pass<PACKED, 128, 128>(tB, wk2 + 3 * s2, 4, wave, lane, d);
  __syncthreads();

  // ---------------- FC epilogue straight from accumulators ----------------
  {
    int col = wave * 16 + (lane & 15);
    float add = ws[1152 + col] + b2[col];
    float p00 = 0.f, p01 = 0.f, p10 = 0.f, p11 = 0.f;
#pragma unroll
    for (int rt = 0; rt < 4; ++rt) {
#pragma unroll
      for (int r = 0; r < 8; ++r) {
        int n = (rt & 1) * 16 + r + ((lane >> 4) << 3);  // node within batch
        float v = d[rt][r] + add;
        v = (v > 0.f) ? v : 0.f;
        int f = n * 128 + col;  // flat = n*128 + h
        if (rt < 2) {
          p00 += v * fc_w[f];
          p01 += v * fc_w[4096 + f];
        } else {
          p10 += v * fc_w[f];
          p11 += v * fc_w[4096 + f];
        }
      }
    }
    red[tid] = p00;
    red[256 + tid] = p01;
    red[512 + tid] = p10;
    red[768 + tid] = p11;
  }
  __syncthreads();
  if (tid < 4) {  // deterministic fixed-order reduction
    int bat = tid >> 1, o = tid & 1;
    float s = fc_b[o];
    const float* pr = red + tid * 256;  // tid = bat*2+o matches layout
    for (int i = 0; i < 256; ++i) s += pr[i];
    out[(size_t)(b0 + bat) * 2 + o] = s;
  }
}

// ---------------------------------------------------------------------------
extern "C" void kernel_launch(void* const* d_in, const int* in_sizes, int n_in,
                              void* d_out, int out_size, void* d_ws,
                              size_t ws_size, hipStream_t stream) {
  const float* x = (const float*)d_in[0];
  const float* adj = (const float*)d_in[1];
  const float* adj_bias = (const float*)d_in[2];
  const float* w1 = (const float*)d_in[3];
  const float* b1 = (const float*)d_in[4];
  const float* w2 = (const float*)d_in[5];
  const float* b2 = (const float*)d_in[6];
  const float* fc_w = (const float*)d_in[7];
  const float* fc_b = (const float*)d_in[8];
  float* ws = (float*)d_ws;
  float* out = (float*)d_out;

  (void)in_sizes; (void)n_in; (void)out_size;

  const bool packed = ws_size >= (size_t)WS_NEED_FLOATS * sizeof(float);
  if (packed) {
    dgcnn_stage0<true><<<1, 256, 0, stream>>>(adj, adj_bias, w1, w2, ws);
    dgcnn_main<true><<<4096, 256, 0, stream>>>(x, w1, b1, w2, b2, fc_w, fc_b,
                                               ws, out);
  } else {
    dgcnn_stage0<false><<<1, 256, 0, stream>>>(adj, adj_bias, w1, w2, ws);
    dgcnn_main<false><<<4096, 256, 0, stream>>>(x, w1, b1, w2, b2, fc_w, fc_b,
                                                ws, out);
  }
}